// SparsePointEncoder_59622736003147
// MI455X (gfx1250) — compile-verified
//
#include <hip/hip_runtime.h>
#include <hip/hip_bf16.h>
#include <math.h>

// ---------------- problem constants ----------------
#define BT      4
#define NPB     65536
#define P_TOT   (BT * NPB)          // 262144 points
#define HC      (1u << 20)          // hash capacity (power of two)
#define IMAXK   0x7fffffff

typedef __attribute__((ext_vector_type(2))) float v2f;
typedef __attribute__((ext_vector_type(8))) float v8f;

__device__ __forceinline__ unsigned hash0(int key) {
    return ((unsigned)key * 2654435761u) & (HC - 1u);
}

__device__ __forceinline__ int encode_key(int b, int qx, int qy, int qz) {
    return ((b * 512 + qx + 256) * 512 + qy + 256) * 128 + qz + 64;
}

// quantize one point (matches reference f32 math)
__device__ __forceinline__ int point_key(const float* __restrict__ xyzi, int b, int n) {
    const float* base = xyzi + (size_t)b * 7 * NPB + n;
    float x = base[0];
    float y = base[(size_t)NPB];
    float z = base[(size_t)2 * NPB];
    float r = fmaxf(sqrtf(x * x + y * y), 1e-6f);
    float w = 5.0f * log1pf(r * 0.2f) / r;      // ALPHA=5, r/ALPHA
    int qx = (int)floorf(x * w / 0.07f);
    int qy = (int)floorf(y * w / 0.07f);
    int qz = (int)floorf(z / 0.1f);
    return encode_key(b, qx, qy, qz);
}

// ---------------- stage 0: init scratch ----------------
__global__ void k_init(int* hk, int* hv, int* nvox, int* cnt, float* vfeat, float* stats) {
    int i = blockIdx.x * blockDim.x + threadIdx.x;
    if ((unsigned)i < HC) { hk[i] = IMAXK; hv[i] = -1; }
    if (i < P_TOT) cnt[i] = 0;
    if (i < 4 * (P_TOT + 1)) vfeat[i] = 0.0f;        // includes the zero row
    if (i < 512) stats[i] = 0.0f;
    if (i == 0) *nvox = 0;
}

// ---------------- stage 1: claim hash slots, assign voxel ids ----------------
__global__ void k_assign(const float* __restrict__ xyzi, const float* __restrict__ vm,
                         int* hk, int* hv, int* nvox, int* vox_key) {
    int p = blockIdx.x * blockDim.x + threadIdx.x;
    if (p >= P_TOT) return;
    if (vm[p] <= 0.5f) return;
    int b = p >> 16, n = p & 65535;
    int key = point_key(xyzi, b, n);
    unsigned h = hash0(key);
    while (true) {
        int prev = atomicCAS(&hk[h], IMAXK, key);
        if (prev == IMAXK) {
            int id = atomicAdd(nvox, 1);
            vox_key[id] = key;
            hv[h] = id;                 // readers are in later kernels
            break;
        }
        if (prev == key) break;
        h = (h + 1) & (HC - 1u);
    }
}

// zero the "missing neighbor" rows h1[nvox], h2[nvox] (vfeat's is zeroed by k_init)
__global__ void k_zerorows(const int* __restrict__ nvoxp, float* h1, float* h2) {
    int nv = *nvoxp;
    int t = threadIdx.x;
    if (t < 32) h1[(size_t)nv * 32 + t] = 0.0f;
    if (t >= 32 && t < 96) h2[(size_t)nv * 64 + (t - 32)] = 0.0f;
}

// ---------------- stage 2: lookup + feature accumulation ----------------
__global__ void k_accum(const float* __restrict__ xyzi, const float* __restrict__ vm,
                        const int* __restrict__ hk, const int* __restrict__ hv,
                        int* inv, int* cnt, float* vfeat) {
    int p = blockIdx.x * blockDim.x + threadIdx.x;
    if (p >= P_TOT) return;
    int b = p >> 16, n = p & 65535;
    if (vm[p] <= 0.5f) { inv[p] = -1; return; }
    int key = point_key(xyzi, b, n);
    unsigned h = hash0(key);
    int id = -1;
    while (true) {
        int k = hk[h];
        if (k == key) { id = hv[h]; break; }
        if (k == IMAXK) break;
        h = (h + 1) & (HC - 1u);
    }
    inv[p] = id;
    if (id >= 0) {
        atomicAdd(&cnt[id], 1);
        const float* fb = xyzi + (size_t)b * 7 * NPB + n + (size_t)3 * NPB;
#pragma unroll
        for (int c = 0; c < 4; c++)
            atomicAdd(&vfeat[(size_t)id * 4 + c], fb[(size_t)c * NPB]);
    }
}

// ---------------- stage 3: finalize voxel feats + build 27-neighbor table ----------------
// Missing neighbors are mapped to row `nvox` (the guaranteed-zero row).
__global__ void k_voxfin(const int* __restrict__ nvoxp, const int* __restrict__ vox_key,
                         const int* __restrict__ hk, const int* __restrict__ hv,
                         const int* __restrict__ cnt, float* vfeat, int* nb) {
    int nvox = *nvoxp;
    int v = blockIdx.x * blockDim.x + threadIdx.x;
    if (v >= nvox) return;
    float s = 1.0f / fmaxf((float)cnt[v], 1.0f);
#pragma unroll
    for (int c = 0; c < 4; c++) vfeat[(size_t)v * 4 + c] *= s;

    int key = vox_key[v];
    int qz = key % 128 - 64; int t = key / 128;
    int qy = t % 512 - 256;  t /= 512;
    int qx = t % 512 - 256;  int b = t / 512;

    for (int k = 0; k < 27; k++) {
        int dx = k / 9 - 1, dy = (k / 3) % 3 - 1, dz = k % 3 - 1;  // reference OFFS order
        int nkey = encode_key(b, qx + dx, qy + dy, qz + dz);
        unsigned h = hash0(nkey);
        int id = -1;
        while (true) {
            int kk = hk[h];
            if (kk == nkey) { id = hv[h]; break; }
            if (kk == IMAXK) break;
            h = (h + 1) & (HC - 1u);
        }
        nb[(size_t)v * 27 + k] = (id < 0) ? nvox : id;
    }
}

// ---------------- weight pre-pack into per-fragment, per-lane float2 ----------------
// Fragment f = (k*(CIN/4)+kk)*(COUT/16)+nt; lane l = hi*16+m holds
// { W[(k*CIN+kk*4+hi*2  )*COUT + nt*16+m],
//   W[(k*CIN+kk*4+hi*2+1)*COUT + nt*16+m] }  -> one coalesced b64 load per fragment.
template <int CIN, int COUT, int NK>
__global__ void k_pack(const float* __restrict__ W, v2f* __restrict__ Wt) {
    int i = blockIdx.x * blockDim.x + threadIdx.x;
    const int total = NK * (CIN / 4) * (COUT / 16) * 32;
    if (i >= total) return;
    int l = i & 31, frag = i >> 5;
    int nt = frag % (COUT / 16); int rest = frag / (COUT / 16);
    int kk = rest % (CIN / 4);   int k = rest / (CIN / 4);
    int m = l & 15, hi = l >> 4;
    int rB = k * CIN + kk * 4 + hi * 2;
    int n = nt * 16 + m;
    v2f t;
    t.x = W[(size_t)rB * COUT + n];
    t.y = W[(size_t)(rB + 1) * COUT + n];
    Wt[i] = t;
}

// ---------------- WMMA sparse/dense conv ----------------
// One wave computes a 32-voxel x COUT super-tile (two 16x16 M-tiles sharing each
// B-fragment) with V_WMMA_F32_16X16X4_F32. All loads are unconditional:
// out-of-range lanes read a clamped (real) voxel's nb entry; the resulting
// garbage lands only in accumulator rows the guarded store discards, and
// missing neighbors point at the guaranteed-zero row (index nvox).
template <int CIN, int COUT, bool GATHER>
__global__ void k_conv(const float* __restrict__ hin, const v2f* __restrict__ Wt,
                       const int* __restrict__ nb, const int* __restrict__ nvoxp,
                       float* __restrict__ hout) {
    const int nvox = *nvoxp;
    const int wave = blockIdx.x * (blockDim.x >> 5) + (threadIdx.x >> 5);
    const int tile = wave << 5;              // 32 voxels per wave
    if (tile >= nvox) return;

    const int l  = threadIdx.x & 31;
    const int m  = l & 15;     // M row (A) / N col (B,C)
    const int hi = l >> 4;     // K-half select

    v8f zero8;
#pragma unroll
    for (int e = 0; e < 8; e++) zero8[e] = 0.0f;
    v8f acc0[COUT / 16], acc1[COUT / 16];
#pragma unroll
    for (int i = 0; i < COUT / 16; i++) { acc0[i] = zero8; acc1[i] = zero8; }

    const int NK = GATHER ? 27 : 1;
    const int vc0 = min(tile + m, nvox - 1);        // clamped: always a real voxel
    const int vc1 = min(tile + 16 + m, nvox - 1);
    const int* nbp0 = nb + (size_t)vc0 * 27;
    const int* nbp1 = nb + (size_t)vc1 * 27;
    const int koff = hi * 2;
    const v2f* wl = Wt + l;

    for (int k = 0; k < NK; k++) {
        int row0, row1;
        if (GATHER) { row0 = nbp0[k]; row1 = nbp1[k]; }   // unconditional b32 loads
        else        { row0 = vc0;     row1 = vc1; }
        const float* ar0 = hin + (size_t)row0 * CIN + koff;
        const float* ar1 = hin + (size_t)row1 * CIN + koff;
#pragma unroll
        for (int kk = 0; kk < CIN / 4; kk++) {
            v2f a0 = *reinterpret_cast<const v2f*>(ar0 + kk * 4);
            v2f a1 = *reinterpret_cast<const v2f*>(ar1 + kk * 4);
            const v2f* wrow = wl + (size_t)(k * (CIN / 4) + kk) * (COUT / 16) * 32;
#pragma unroll
            for (int nt = 0; nt < COUT / 16; nt++) {
                v2f bf = wrow[(size_t)nt * 32];
                acc0[nt] = __builtin_amdgcn_wmma_f32_16x16x4_f32(
                    false, a0, false, bf, (short)0, acc0[nt], false, false);
                acc1[nt] = __builtin_amdgcn_wmma_f32_16x16x4_f32(
                    false, a1, false, bf, (short)0, acc1[nt], false, false);
            }
        }
    }

    // store: lane l element e -> (M = e + hi*8 [+16], N = nt*16 + m)
    // All of a lane's stores share one base address; offsets are compile-time
    // immediates, so the full-tile fast path is a pure store clause.
    float* ob = hout + (size_t)(tile + hi * 8) * COUT + m;
    if (tile + 32 <= nvox) {
#pragma unroll
        for (int nt = 0; nt < COUT / 16; nt++) {
#pragma unroll
            for (int e = 0; e < 8; e++) {
                ob[e * COUT + nt * 16]        = acc0[nt][e];
                ob[(16 + e) * COUT + nt * 16] = acc1[nt][e];
            }
        }
    } else {
#pragma unroll
        for (int nt = 0; nt < COUT / 16; nt++) {
#pragma unroll
            for (int e = 0; e < 8; e++) {
                int r0 = tile + e + hi * 8;
                if (r0 < nvox)      ob[e * COUT + nt * 16]        = acc0[nt][e];
                if (r0 + 16 < nvox) ob[(16 + e) * COUT + nt * 16] = acc1[nt][e];
            }
        }
    }
}

// ---------------- BN stats: per-channel sum(x), sum(x^2) ----------------
template <int C>
__global__ void k_bnstats(const float* __restrict__ x, const int* __restrict__ nvoxp,
                          float* __restrict__ sums) {
    const int nvox = *nvoxp;
    int tid = blockIdx.x * blockDim.x + threadIdx.x;
    int c = tid & (C - 1);
    int row = tid / C;
    int rstep = (gridDim.x * blockDim.x) / C;
    float s = 0.0f, ss = 0.0f;
    for (; row < nvox; row += rstep) {
        float t = x[(size_t)row * C + c];
        s += t; ss += t * t;
    }
    atomicAdd(&sums[c], s);
    atomicAdd(&sums[C + c], ss);
}

// ---------------- BN apply (in place) + ReLU ----------------
template <int C>
__global__ void k_bnapply(float* __restrict__ x, const float* __restrict__ gamma,
                          const float* __restrict__ beta, const float* __restrict__ sums,
                          const int* __restrict__ nvoxp) {
    const int nvox = *nvoxp;
    int tid = blockIdx.x * blockDim.x + threadIdx.x;
    int row = tid / C;
    if (row >= nvox) return;
    int c = tid & (C - 1);
    float nf = fmaxf((float)nvox, 1.0f);
    float mu = sums[c] / nf;
    float var = fmaxf(sums[C + c] / nf - mu * mu, 0.0f);
    float sc = gamma[c] * rsqrtf(var + 1e-5f);
    float y = (x[tid] - mu) * sc + beta[c];
    x[tid] = fmaxf(y, 0.0f);
}

// ---------------- final scatter to (BT, 64, N, 1) ----------------
__global__ void k_scatter(const float* __restrict__ h3, const int* __restrict__ inv,
                          float* __restrict__ out) {
    int p = blockIdx.x * blockDim.x + threadIdx.x;
    if (p >= P_TOT) return;
    int id = inv[p];
    int b = p >> 16, n = p & 65535;
    float* ob = out + (size_t)b * 64 * NPB + n;
    if (id >= 0) {
        const float* hr = h3 + (size_t)id * 64;
#pragma unroll
        for (int c = 0; c < 64; c++) ob[(size_t)c * NPB] = hr[c];
    } else {
#pragma unroll
        for (int c = 0; c < 64; c++) ob[(size_t)c * NPB] = 0.0f;
    }
}

// ---------------- launcher ----------------
extern "C" void kernel_launch(void* const* d_in, const int* in_sizes, int n_in,
                              void* d_out, int out_size, void* d_ws, size_t ws_size,
                              hipStream_t stream) {
    (void)in_sizes; (void)n_in; (void)out_size; (void)ws_size;
    const float* xyzi = (const float*)d_in[0];
    const float* vm   = (const float*)d_in[1];
    const float* W1   = (const float*)d_in[2];
    const float* g1   = (const float*)d_in[3];
    const float* b1   = (const float*)d_in[4];
    const float* W2   = (const float*)d_in[5];
    const float* g2   = (const float*)d_in[6];
    const float* b2   = (const float*)d_in[7];
    const float* W3   = (const float*)d_in[8];
    const float* g3   = (const float*)d_in[9];
    const float* b3   = (const float*)d_in[10];
    float* out = (float*)d_out;

    char* ws = (char*)d_ws;
    size_t off = 0;
    auto alloc = [&](size_t bytes) -> void* {
        void* p = ws + off;
        off = (off + bytes + 255) & ~(size_t)255;
        return p;
    };
    int*   hk      = (int*)alloc((size_t)HC * 4);
    int*   hv      = (int*)alloc((size_t)HC * 4);
    int*   nvox    = (int*)alloc(256);
    int*   vox_key = (int*)alloc((size_t)P_TOT * 4);
    int*   inv     = (int*)alloc((size_t)P_TOT * 4);
    int*   cnt     = (int*)alloc((size_t)P_TOT * 4);
    float* vfeat   = (float*)alloc((size_t)(P_TOT + 1) * 4 * 4);   // +zero row
    int*   nb      = (int*)alloc((size_t)P_TOT * 27 * 4);
    float* h1      = (float*)alloc((size_t)(P_TOT + 1) * 32 * 4);  // +zero row
    float* h2      = (float*)alloc((size_t)(P_TOT + 1) * 64 * 4);  // +zero row
    float* h3      = (float*)alloc((size_t)P_TOT * 64 * 4);
    v2f*   Wt1     = (v2f*)alloc((size_t)27 * 1 * 2 * 32 * 8);
    v2f*   Wt2     = (v2f*)alloc((size_t)27 * 8 * 4 * 32 * 8);
    v2f*   Wt3     = (v2f*)alloc((size_t)1 * 16 * 4 * 32 * 8);
    float* stats   = (float*)alloc(512 * 4);

    k_init<<<HC / 256 + 1, 256, 0, stream>>>(hk, hv, nvox, cnt, vfeat, stats);
    k_pack<4, 32, 27><<<7, 256, 0, stream>>>(W1, Wt1);
    k_pack<32, 64, 27><<<108, 256, 0, stream>>>(W2, Wt2);
    k_pack<64, 64, 1><<<8, 256, 0, stream>>>(W3, Wt3);

    k_assign<<<P_TOT / 256, 256, 0, stream>>>(xyzi, vm, hk, hv, nvox, vox_key);
    k_zerorows<<<1, 128, 0, stream>>>(nvox, h1, h2);
    k_accum<<<P_TOT / 256, 256, 0, stream>>>(xyzi, vm, hk, hv, inv, cnt, vfeat);
    k_voxfin<<<P_TOT / 256, 256, 0, stream>>>(nvox, vox_key, hk, hv, cnt, vfeat, nb);

    // layer 1: 4 -> 32, 27 offsets
    k_conv<4, 32, true><<<1024, 256, 0, stream>>>(vfeat, Wt1, nb, nvox, h1);
    k_bnstats<32><<<256, 256, 0, stream>>>(h1, nvox, stats);
    k_bnapply<32><<<(P_TOT * 32) / 256, 256, 0, stream>>>(h1, g1, b1, stats, nvox);

    // layer 2: 32 -> 64, 27 offsets (dominant FLOPs)
    k_conv<32, 64, true><<<1024, 256, 0, stream>>>(h1, Wt2, nb, nvox, h2);
    k_bnstats<64><<<256, 256, 0, stream>>>(h2, nvox, stats + 64);
    k_bnapply<64><<<(P_TOT * 64) / 256, 256, 0, stream>>>(h2, g2, b2, stats + 64, nvox);

    // layer 3: dense 64 -> 64
    k_conv<64, 64, false><<<1024, 256, 0, stream>>>(h2, Wt3, nullptr, nvox, h3);
    k_bnstats<64><<<256, 256, 0, stream>>>(h3, nvox, stats + 192);
    k_bnapply<64><<<(P_TOT * 64) / 256, 256, 0, stream>>>(h3, g3, b3, stats + 192, nvox);

    k_scatter<<<P_TOT / 256, 256, 0, stream>>>(h3, inv, out);
}